// GCN_10866267259524
// MI455X (gfx1250) — compile-verified
//
#include <hip/hip_runtime.h>

#define N_NODES  50000
#define N_EDGES  800000
#define F        128
#define OUTF     64
#define N_GRAPHS 128

typedef float v2f __attribute__((ext_vector_type(2)));
typedef float v8f __attribute__((ext_vector_type(8)));

// ---------------------------------------------------------------- utilities
__global__ void fill_zero(float* __restrict__ p, int n) {
    int i = blockIdx.x * blockDim.x + threadIdx.x;
    if (i < n) p[i] = 0.0f;
}

// deg[d] += 1 for each edge (dst row of edge_index)
__global__ void deg_kernel(const long long* __restrict__ ei, float* __restrict__ deg) {
    int e = blockIdx.x * blockDim.x + threadIdx.x;
    if (e < N_EDGES) {
        int d = (int)ei[N_EDGES + e];
        atomicAdd(&deg[d], 1.0f);
    }
}

// in-place: deg -> rsqrt(deg + 1)   (+1 self loop)
__global__ void dinv_kernel(float* __restrict__ deg) {
    int i = blockIdx.x * blockDim.x + threadIdx.x;
    if (i < N_NODES) deg[i] = rsqrtf(deg[i] + 1.0f);
}

// ------------------------------------------------ dense GEMM: H = X @ W (128x128)
// One wave (32 lanes) computes a 16-row x 128-col strip using v_wmma_f32_16x16x4_f32.
// W is staged in LDS pre-swizzled into B-fragment order:
//   frag element (s, t, lane, j):  k = 4*s + 2*(lane>=16) + j,  n = 16*t + (lane&15)
// A fragment per lane: X[row0 + (lane&15)][4*s + 2*(lane>=16) + {0,1}]  (contiguous float2)
// C/D layout: acc[t] vgpr r -> row = row0 + r + 8*(lane>=16), col = 16*t + (lane&15)
__global__ __launch_bounds__(256) void gemm_x128_wmma(const float* __restrict__ X,
                                                      const float* __restrict__ W,
                                                      float* __restrict__ H,
                                                      int n_rows) {
    __shared__ float wlds[32 * 8 * 32 * 2];   // 16384 floats = 64 KB

    const int tid = threadIdx.x;
    // cooperative fill of W fragments (8192 float2 elements)
    for (int idx2 = tid; idx2 < 32 * 8 * 32; idx2 += 256) {
        int lane = idx2 & 31;
        int t    = (idx2 >> 5) & 7;
        int s    = idx2 >> 8;
        int k    = 4 * s + 2 * (lane >> 4);
        int n    = 16 * t + (lane & 15);
        float2 w2 = make_float2(W[k * F + n], W[(k + 1) * F + n]);
        *(float2*)&wlds[idx2 * 2] = w2;
    }
    __syncthreads();

    const int wave = tid >> 5;
    const int lane = tid & 31;
    const int row0 = (blockIdx.x * 8 + wave) * 16;
    // clamp load row so EXEC stays all-ones through the WMMA sequence
    const int rload = min(row0 + (lane & 15), n_rows - 1);
    const float* xrow = X + (size_t)rload * F;
    const int khalf = 2 * (lane >> 4);

    v8f acc[8] = {};   // zero accumulators

    for (int s = 0; s < 32; ++s) {
        int k = 4 * s + khalf;
        v2f a = *(const v2f*)(xrow + k);
        const float* wp = &wlds[s * 512 + lane * 2];
#pragma unroll
        for (int t = 0; t < 8; ++t) {
            v2f b = *(const v2f*)(wp + t * 64);
            acc[t] = __builtin_amdgcn_wmma_f32_16x16x4_f32(
                false, a, false, b, (short)0, acc[t], false, false);
        }
    }

    // store (bounds-checked; all WMMAs already done)
    const int col0  = lane & 15;
    const int mhalf = (lane >> 4) * 8;
#pragma unroll
    for (int r = 0; r < 8; ++r) {
        int row = row0 + mhalf + r;
        if (row < n_rows) {
            float* hrow = H + (size_t)row * F + col0;
#pragma unroll
            for (int t = 0; t < 8; ++t) hrow[t * 16] = acc[t][r];
        }
    }
}

// ---------------------------------------- edge scatter: agg[dst] += h[src] * norm
// one lane per 4 features: 32 lanes cover the 128-wide feature row of an edge
__global__ void scatter_kernel(const long long* __restrict__ ei,
                               const float* __restrict__ h,
                               const float* __restrict__ dinv,
                               float* __restrict__ agg) {
    long long gid = (long long)blockIdx.x * blockDim.x + threadIdx.x;
    int e = (int)(gid >> 5);
    if (e >= N_EDGES) return;
    int lane = (int)(gid & 31);
    int s = (int)ei[e];
    int d = (int)ei[N_EDGES + e];
    float nrm = dinv[s] * dinv[d];
    const float4 hv = *(const float4*)(h + (size_t)s * F + lane * 4);
    float* ap = agg + (size_t)d * F + lane * 4;
    atomicAdd(ap + 0, hv.x * nrm);
    atomicAdd(ap + 1, hv.y * nrm);
    atomicAdd(ap + 2, hv.z * nrm);
    atomicAdd(ap + 3, hv.w * nrm);
}

// ------------------- finalize: agg = [relu](agg + h * dinv^2 + bias), in place
__global__ void finalize_kernel(float* __restrict__ agg,
                                const float* __restrict__ h,
                                const float* __restrict__ dinv,
                                const float* __restrict__ bias,
                                int do_relu) {
    int idx = blockIdx.x * blockDim.x + threadIdx.x;
    if (idx >= N_NODES * F) return;
    int i = idx >> 7;
    int f = idx & 127;
    float di = dinv[i];
    float v = agg[idx] + h[idx] * di * di + bias[f];
    if (do_relu) v = fmaxf(v, 0.0f);
    agg[idx] = v;
}

// ------------------------------- global mean pool (accumulate sums and counts)
__global__ void pool_kernel(const float* __restrict__ feat,
                            const long long* __restrict__ batch,
                            float* __restrict__ gsum,
                            float* __restrict__ gcnt) {
    int idx = blockIdx.x * blockDim.x + threadIdx.x;
    if (idx >= N_NODES * F) return;
    int i = idx >> 7;
    int f = idx & 127;
    int g = (int)batch[i];
    atomicAdd(&gsum[g * F + f], feat[idx]);
    if (f == 0) atomicAdd(&gcnt[g], 1.0f);
}

// ------------------------------------- head: out = (gsum/cnt) @ W_lin + b_lin
__global__ void head_kernel(const float* __restrict__ gsum,
                            const float* __restrict__ gcnt,
                            const float* __restrict__ Wl,
                            const float* __restrict__ bl,
                            float* __restrict__ out) {
    int idx = blockIdx.x * blockDim.x + threadIdx.x;
    if (idx >= N_GRAPHS * OUTF) return;
    int g = idx / OUTF;
    int o = idx % OUTF;
    float inv = 1.0f / fmaxf(gcnt[g], 1.0f);
    const float* gr = gsum + g * F;
    float acc = bl[o];
#pragma unroll 4
    for (int k = 0; k < F; ++k) acc = fmaf(gr[k] * inv, Wl[k * OUTF + o], acc);
    out[idx] = acc;
}

// ---------------------------------------------------------------------------
extern "C" void kernel_launch(void* const* d_in, const int* in_sizes, int n_in,
                              void* d_out, int out_size, void* d_ws, size_t ws_size,
                              hipStream_t stream) {
    const float*     x     = (const float*)d_in[0];
    const long long* ei    = (const long long*)d_in[1];
    const long long* batch = (const long long*)d_in[2];
    const float* W1 = (const float*)d_in[3];
    const float* b1 = (const float*)d_in[4];
    const float* W2 = (const float*)d_in[5];
    const float* b2 = (const float*)d_in[6];
    const float* W3 = (const float*)d_in[7];
    const float* b3 = (const float*)d_in[8];
    const float* Wl = (const float*)d_in[9];
    const float* bl = (const float*)d_in[10];
    float* out = (float*)d_out;

    char* ws = (char*)d_ws;
    const size_t NFB = (size_t)N_NODES * F * sizeof(float);     // 25.6 MB
    float* h    = (float*)ws;
    float* agg  = (float*)(ws + NFB);
    float* dinv = (float*)(ws + 2 * NFB);
    size_t dsz  = ((size_t)N_NODES * sizeof(float) + 255) & ~(size_t)255;
    float* gsum = (float*)(ws + 2 * NFB + dsz);
    float* gcnt = gsum + N_GRAPHS * F;

    const int T = 256;
    const int nelem = N_NODES * F;

    // degrees -> dinv (same for all layers: unit edge weights + self loop)
    fill_zero<<<(N_NODES + T - 1) / T, T, 0, stream>>>(dinv, N_NODES);
    deg_kernel<<<(N_EDGES + T - 1) / T, T, 0, stream>>>(ei, dinv);
    dinv_kernel<<<(N_NODES + T - 1) / T, T, 0, stream>>>(dinv);

    const int strips      = (N_NODES + 15) / 16;   // 3125
    const int gemm_blocks = (strips + 7) / 8;      // 8 waves per block

    const float* feat = x;
    const float* Ws[3] = {W1, W2, W3};
    const float* bs[3] = {b1, b2, b3};
    for (int l = 0; l < 3; ++l) {
        gemm_x128_wmma<<<gemm_blocks, 256, 0, stream>>>(feat, Ws[l], h, N_NODES);
        fill_zero<<<(nelem + T - 1) / T, T, 0, stream>>>(agg, nelem);
        long long scatter_work = (long long)N_EDGES * 32;
        scatter_kernel<<<(int)((scatter_work + T - 1) / T), T, 0, stream>>>(ei, h, dinv, agg);
        finalize_kernel<<<(nelem + T - 1) / T, T, 0, stream>>>(agg, h, dinv, bs[l], l < 2);
        feat = agg;   // agg buffer becomes next layer's features
    }

    fill_zero<<<(N_GRAPHS * F + N_GRAPHS + T - 1) / T, T, 0, stream>>>(gsum, N_GRAPHS * F + N_GRAPHS);
    pool_kernel<<<(nelem + T - 1) / T, T, 0, stream>>>(feat, batch, gsum, gcnt);
    head_kernel<<<(N_GRAPHS * OUTF + T - 1) / T, T, 0, stream>>>(gsum, gcnt, Wl, bl, out);
}